// SGC_29506425323818
// MI455X (gfx1250) — compile-verified
//
#include <hip/hip_runtime.h>

#define N_NODES 50000
#define N_EDGES 800000
#define IN_FEATS 128
#define NUM_CLASSES 64

typedef float v2f __attribute__((ext_vector_type(2)));
typedef float v8f __attribute__((ext_vector_type(8)));

// ---------------------------------------------------------------------------
// Phase 0: zero degree + agg scratch (ws is poisoned 0xAA by harness)
// ---------------------------------------------------------------------------
__global__ void sgc_zero_kernel(float* __restrict__ p, int n) {
  int i = blockIdx.x * blockDim.x + threadIdx.x;
  if (i < n) p[i] = 0.0f;
}

// ---------------------------------------------------------------------------
// Phase 1: in-degree via f32 atomics (L2-resident, 50K-node array)
// ---------------------------------------------------------------------------
__global__ void sgc_degree_kernel(const int* __restrict__ dst,
                                  float* __restrict__ deg) {
  int e = blockIdx.x * blockDim.x + threadIdx.x;
  if (e < N_EDGES) atomicAdd(&deg[dst[e]], 1.0f);
}

// ---------------------------------------------------------------------------
// Phase 2: norm = rsqrt(max(deg,1)) in-place
// ---------------------------------------------------------------------------
__global__ void sgc_norm_kernel(float* __restrict__ deg) {
  int i = blockIdx.x * blockDim.x + threadIdx.x;
  if (i < N_NODES) deg[i] = rsqrtf(fmaxf(deg[i], 1.0f));
}

// ---------------------------------------------------------------------------
// Phase 3: edge scatter. One wave32 per edge; lane l handles features
// [4l, 4l+3] as a float4 -> coalesced 512B gather + coalesced atomic adds.
// agg (25.6MB) sits in the 192MB L2, so all atomics resolve on-chip.
// ---------------------------------------------------------------------------
__global__ void sgc_scatter_kernel(const float* __restrict__ in_feat,
                                   const int* __restrict__ src,
                                   const int* __restrict__ dst,
                                   const float* __restrict__ norm,
                                   float* __restrict__ agg) {
  int tid  = blockIdx.x * blockDim.x + threadIdx.x;
  int e    = tid >> 5;
  int lane = tid & 31;
  if (e >= N_EDGES) return;
  int s = src[e];
  int d = dst[e];
  float ns = norm[s];
  float4 v = *(const float4*)(in_feat + (size_t)s * IN_FEATS + lane * 4);
  float* o = agg + (size_t)d * IN_FEATS + lane * 4;
  atomicAdd(o + 0, v.x * ns);
  atomicAdd(o + 1, v.y * ns);
  atomicAdd(o + 2, v.z * ns);
  atomicAdd(o + 3, v.w * ns);
}

// ---------------------------------------------------------------------------
// Phase 4: out = (agg * norm[:,None]) @ W + b  via V_WMMA_F32_16X16X4_F32.
// Block = 128 threads = 4 wave32. Block owns 16 nodes (M); wave w owns
// classes [16w,16w+16) (N). A tile (16x128, norm-scaled) staged in LDS,
// padded to 132 floats/row to spread the K-column reads across banks.
// K=128 -> 32 chained wmma ops, fp32 end to end (matches fp32 reference).
// ---------------------------------------------------------------------------
__global__ void __launch_bounds__(128)
sgc_gemm_kernel(const float* __restrict__ agg,
                const float* __restrict__ norm,
                const float* __restrict__ W,
                const float* __restrict__ bias,
                float* __restrict__ out) {
  __shared__ float As[16][132];

  const int tid   = threadIdx.x;
  const int node0 = blockIdx.x * 16;   // 50000 = 16 * 3125, exact

  // Cooperative stage of norm-scaled A tile: 16 rows x 32 float4s.
  for (int i = tid; i < 16 * 32; i += 128) {
    int r  = i >> 5;
    int c4 = (i & 31) * 4;
    float nn = norm[node0 + r];
    float4 v = *(const float4*)(agg + (size_t)(node0 + r) * IN_FEATS + c4);
    As[r][c4 + 0] = v.x * nn;
    As[r][c4 + 1] = v.y * nn;
    As[r][c4 + 2] = v.z * nn;
    As[r][c4 + 3] = v.w * nn;
  }
  __syncthreads();

  const int wave   = tid >> 5;
  const int lane   = tid & 31;
  const int class0 = wave * 16;        // 64 = 16 * 4, exact
  const int half   = lane >> 4;        // 0: K={0,1}, 1: K={2,3}
  const int mn     = lane & 15;        // row for A, col for B/D
  const int kb     = half * 2;

  v8f c = {};
  for (int k0 = 0; k0 < IN_FEATS; k0 += 4) {
    v2f a, bm;
    a.x  = As[mn][k0 + kb];
    a.y  = As[mn][k0 + kb + 1];
    bm.x = W[(k0 + kb)     * NUM_CLASSES + class0 + mn];
    bm.y = W[(k0 + kb + 1) * NUM_CLASSES + class0 + mn];
    // D = A(16x4) x B(4x16) + C, fp32
    c = __builtin_amdgcn_wmma_f32_16x16x4_f32(
        /*neg_a=*/false, a, /*neg_b=*/false, bm,
        /*c_mod=*/(short)0, c, /*reuse_a=*/false, /*reuse_b=*/false);
  }

  const float bb    = bias[class0 + mn];
  const int   mbase = half * 8;        // C/D layout: lanes 16-31 hold M+8
  for (int v = 0; v < 8; ++v) {
    out[(size_t)(node0 + mbase + v) * NUM_CLASSES + class0 + mn] = c[v] + bb;
  }
}

// ---------------------------------------------------------------------------
extern "C" void kernel_launch(void* const* d_in, const int* in_sizes, int n_in,
                              void* d_out, int out_size, void* d_ws, size_t ws_size,
                              hipStream_t stream) {
  const float* in_feat = (const float*)d_in[0];
  const float* W       = (const float*)d_in[1];
  const float* b       = (const float*)d_in[2];
  const int*   src     = (const int*)d_in[3];
  const int*   dst     = (const int*)d_in[4];
  float*       out     = (float*)d_out;

  // ws layout: [deg/norm : N_NODES floats][agg : N_NODES*IN_FEATS floats]
  float* deg = (float*)d_ws;
  float* agg = deg + N_NODES;

  const int zn = N_NODES + N_NODES * IN_FEATS;
  sgc_zero_kernel<<<(zn + 255) / 256, 256, 0, stream>>>(deg, zn);

  sgc_degree_kernel<<<(N_EDGES + 255) / 256, 256, 0, stream>>>(dst, deg);
  sgc_norm_kernel<<<(N_NODES + 255) / 256, 256, 0, stream>>>(deg);

  const long long scatter_threads = (long long)N_EDGES * 32;
  sgc_scatter_kernel<<<(int)((scatter_threads + 255) / 256), 256, 0, stream>>>(
      in_feat, src, dst, deg, agg);

  sgc_gemm_kernel<<<N_NODES / 16, 128, 0, stream>>>(agg, deg, W, b, out);
}